// DPhysicsEngine_75376676044839
// MI455X (gfx1250) — compile-verified
//
#include <hip/hip_runtime.h>
#include <math.h>

// DPhysicsEngine step for MI455X (gfx1250), one workgroup per batch element.
// Memory-bound gather workload: single pass over terrain grids, per-point
// state staged in LDS, deterministic tree reductions, f32 WMMA for the
// rigid-frame transform gpts = x + R*lp (homogeneous 16x4 @ 4x16 tiles).

#define NPTS 1024
#define DG   128
#define NJ   4
#define BLK  256
#define NWAVES (BLK / 32)

#define MASS_C   50.0f
#define GRAV_C   9.81f
#define DT_C     0.01f
#define TLIM_C   200.0f
#define KST_C    5000.0f
#define KFR_C    0.8f
#define MAXC_C   3.2f
#define JVL_C    1.0f
#define JL_C     1.5f
#define KD_C     1000.0f   // sqrt(4*MASS*KSTIFF)*ALPHA

typedef float v2f __attribute__((ext_vector_type(2)));
typedef float v8f __attribute__((ext_vector_type(8)));

__device__ __forceinline__ float wsum32(float v) {
#pragma unroll
  for (int o = 16; o > 0; o >>= 1) v += __shfl_down(v, o, 32);
  return v;
}
__device__ __forceinline__ float clampf(float v, float lo, float hi) {
  return fminf(fmaxf(v, lo), hi);
}

__global__ __launch_bounds__(BLK) void dphys_step(
    const float* __restrict__ x, const float* __restrict__ xd,
    const float* __restrict__ R, const float* __restrict__ lp,
    const float* __restrict__ omega, const float* __restrict__ thetas,
    const float* __restrict__ controls, const float* __restrict__ zg,
    const float* __restrict__ gg, const float* __restrict__ pmass,
    const float* __restrict__ dpm, const float* __restrict__ body,
    const float* __restrict__ jpos, float* __restrict__ out, int Bn) {
  __shared__ float sgp[NPTS * 3];            // global points
  __shared__ float sdh[NPTS];                // masked penetration depth
  __shared__ float snx[NPTS], sny[NPTS], snz[NPTS];  // surface normal
  __shared__ float sic[NPTS];                // in-contact flag
  __shared__ float spart[NWAVES][18];        // per-wave reduction partials
  __shared__ float sbc[4];                   // cog.xyz, 1/contact_denom

  const int tid  = threadIdx.x;
  const int lane = tid & 31;
  const int wv   = tid >> 5;
  const int b    = blockIdx.x;
  const size_t NB = (size_t)Bn;

  const float*  Rb  = R        + (size_t)b * 9;
  const float*  xb  = x        + (size_t)b * 3;
  const float*  xdb = xd       + (size_t)b * 3;
  const float*  omb = omega    + (size_t)b * 3;
  const float*  ctb = controls + (size_t)b * (2 * NJ);
  const float*  lpb = lp       + (size_t)b * NPTS * 3;
  const float*  zgb = zg       + (size_t)b * DG * DG;
  const float2* ggb = (const float2*)(gg + (size_t)b * DG * DG * 2);

  // flat output offsets (tuple concatenation order)
  const size_t o_x   = 0;
  const size_t o_xd  = 3 * NB;
  const size_t o_R   = 6 * NB;
  const size_t o_lp  = 15 * NB;
  const size_t o_om  = 15 * NB + 3 * NB * (size_t)NPTS;
  const size_t o_th  = 18 * NB + 3 * NB * (size_t)NPTS;
  const size_t o_xdd = 22 * NB + 3 * NB * (size_t)NPTS;
  const size_t o_od  = 25 * NB + 3 * NB * (size_t)NPTS;
  const size_t o_act = 28 * NB + 3 * NB * (size_t)NPTS;

  // ---------------- Stage 1: gpts = x + R*lp via V_WMMA_F32_16X16X4_F32 ----
  // A(16x4) = [lx ly lz 1] per point row; B(4x16) cols j<3 = [R[j][0..2], x[j]]
  // D(16x16): D[m][j] = gpt_j(point m) for j<3.
#if defined(__gfx1250__)
  {
    const int  jc = lane & 15;
    const bool hi = lane >= 16;
    v2f bmat;
    if (jc < 3) {
      if (!hi) { bmat[0] = Rb[jc * 3 + 0]; bmat[1] = Rb[jc * 3 + 1]; } // K=0,1
      else     { bmat[0] = Rb[jc * 3 + 2]; bmat[1] = xb[jc];        } // K=2,3
    } else { bmat[0] = 0.0f; bmat[1] = 0.0f; }
    const int PPW = NPTS / NWAVES;  // points per wave (128)
    for (int t = 0; t < PPW / 16; ++t) {
      const int pbase = wv * PPW + t * 16;
      const float* q = lpb + (size_t)(pbase + jc) * 3;
      v2f amat;
      if (!hi) { amat[0] = q[0]; amat[1] = q[1]; }   // K=0,1 : lx, ly
      else     { amat[0] = q[2]; amat[1] = 1.0f; }   // K=2,3 : lz, 1
      v8f cz = {0.f, 0.f, 0.f, 0.f, 0.f, 0.f, 0.f, 0.f};
      v8f dm = __builtin_amdgcn_wmma_f32_16x16x4_f32(false, amat, false, bmat,
                                                     (short)0, cz, false, false);
      union { v8f v; float f[8]; } u;
      u.v = dm;
      if (jc < 3) {
        const int mb = pbase + (hi ? 8 : 0);  // D layout: M = r + 8*(lane>=16)
#pragma unroll
        for (int r = 0; r < 8; ++r) sgp[(mb + r) * 3 + jc] = u.f[r];
      }
    }
  }
#else
  for (int n = tid; n < NPTS; n += BLK) {
    float lx = lpb[n * 3 + 0], ly = lpb[n * 3 + 1], lz = lpb[n * 3 + 2];
    sgp[n * 3 + 0] = xb[0] + Rb[0] * lx + Rb[1] * ly + Rb[2] * lz;
    sgp[n * 3 + 1] = xb[1] + Rb[3] * lx + Rb[4] * ly + Rb[5] * lz;
    sgp[n * 3 + 2] = xb[2] + Rb[6] * lx + Rb[7] * ly + Rb[8] * lz;
  }
#endif
  __syncthreads();

  // ---------------- Stage 2: terrain lookups, contact, CoG partials --------
  {
    float c0 = 0.f, c1 = 0.f, c2 = 0.f, cm = 0.f, cc = 0.f;
    for (int n = tid; n < NPTS; n += BLK) {
      float gx = sgp[n * 3 + 0], gy = sgp[n * 3 + 1], gz = sgp[n * 3 + 2];
      float pmn = pmass[n];
      c0 += pmn * gx; c1 += pmn * gy; c2 += pmn * gz; cm += pmn;

      float u = (gx + MAXC_C) / (2.0f * MAXC_C) * (float)(DG - 1);
      float v = (gy + MAXC_C) / (2.0f * MAXC_C) * (float)(DG - 1);
      u = clampf(u, 0.0f, (float)(DG - 1));
      v = clampf(v, 0.0f, (float)(DG - 1));
      int x0 = (int)floorf(u), y0 = (int)floorf(v);
      int x1 = min(x0 + 1, DG - 1), y1 = min(y0 + 1, DG - 1);
      float wx = u - (float)x0, wy = v - (float)y0;
      int i00 = y0 * DG + x0, i01 = y0 * DG + x1;
      int i10 = y1 * DG + x0, i11 = y1 * DG + x1;

      float z00 = zgb[i00], z01 = zgb[i01], z10 = zgb[i10], z11 = zgb[i11];
      float top = z00 * (1.0f - wx) + z01 * wx;
      float bot = z10 * (1.0f - wx) + z11 * wx;
      float zv  = top * (1.0f - wy) + bot * wy;

      float dh = gz - zv;
      bool on = (gx >= -MAXC_C) && (gx <= MAXC_C) &&
                (gy >= -MAXC_C) && (gy <= MAXC_C);
      float ic = ((dh <= 0.0f) && on) ? 1.0f : 0.0f;
      dh *= ic;
      cc += ic;

      float2 g00 = ggb[i00], g01 = ggb[i01], g10 = ggb[i10], g11 = ggb[i11];
      float gxv = (g00.x * (1.0f - wx) + g01.x * wx) * (1.0f - wy) +
                  (g10.x * (1.0f - wx) + g11.x * wx) * wy;
      float gyv = (g00.y * (1.0f - wx) + g01.y * wx) * (1.0f - wy) +
                  (g10.y * (1.0f - wx) + g11.y * wx) * wy;
      float nx0 = -gxv, ny0 = -gyv;
      float inv = 1.0f / sqrtf(nx0 * nx0 + ny0 * ny0 + 1.0f);
      snx[n] = nx0 * inv; sny[n] = ny0 * inv; snz[n] = inv;
      sdh[n] = dh; sic[n] = ic;
    }
    c0 = wsum32(c0); c1 = wsum32(c1); c2 = wsum32(c2);
    cm = wsum32(cm); cc = wsum32(cc);
    if (lane == 0) {
      spart[wv][0] = c0; spart[wv][1] = c1; spart[wv][2] = c2;
      spart[wv][3] = cm; spart[wv][4] = cc;
    }
  }
  __syncthreads();

  // ---------------- Stage 3: finalize CoG / contact denominator ------------
  if (tid == 0) {
    float s0 = 0.f, s1 = 0.f, s2 = 0.f, s3 = 0.f, s4 = 0.f;
#pragma unroll
    for (int w = 0; w < NWAVES; ++w) {
      s0 += spart[w][0]; s1 += spart[w][1]; s2 += spart[w][2];
      s3 += spart[w][3]; s4 += spart[w][4];
    }
    sbc[0] = s0 / s3; sbc[1] = s1 / s3; sbc[2] = s2 / s3;
    sbc[3] = 1.0f / fmaxf(s4, 1.0f);
  }
  __syncthreads();

  // ---------------- Stage 4: forces, act, new_lp, inertia/torque partials --
  {
    const float cogx = sbc[0], cogy = sbc[1], cogz = sbc[2], invden = sbc[3];
    const float xdx = xdb[0], xdy = xdb[1], xdz = xdb[2];
    const float omx = omb[0], omy = omb[1], omz = omb[2];
    float thx = Rb[0], thy = Rb[3], thz = Rb[6];   // column 0 of R
    {
      float tn = fmaxf(sqrtf(thx * thx + thy * thy + thz * thz), 1e-8f);
      thx /= tn; thy /= tn; thz /= tn;
    }
    float ctrl[NJ], dth[NJ], cj[NJ], sj[NJ], jpx[NJ], jpy[NJ], jpz[NJ];
#pragma unroll
    for (int i = 0; i < NJ; ++i) {
      ctrl[i] = ctb[i];
      dth[i] = clampf(ctb[NJ + i], -JVL_C, JVL_C) * DT_C;
      cj[i] = cosf(dth[i]); sj[i] = sinf(dth[i]);
      jpx[i] = jpos[i * 3 + 0]; jpy[i] = jpos[i * 3 + 1]; jpz[i] = jpos[i * 3 + 2];
    }
    float pr2 = 0.f, sxx = 0.f, syy = 0.f, szz = 0.f, sxy = 0.f, sxz = 0.f,
          syz = 0.f, tqx = 0.f, tqy = 0.f, tqz = 0.f, fsx = 0.f, fsy = 0.f,
          fsz = 0.f;
    for (int n = tid; n < NPTS; n += BLK) {
      float cx = sgp[n * 3 + 0] - cogx;
      float cy = sgp[n * 3 + 1] - cogy;
      float cz = sgp[n * 3 + 2] - cogz;
      float pmn = pmass[n];
      float r2 = cx * cx + cy * cy + cz * cz;
      pr2 += pmn * r2;
      sxx += pmn * cx * cx; syy += pmn * cy * cy; szz += pmn * cz * cz;
      sxy += pmn * cx * cy; sxz += pmn * cx * cz; syz += pmn * cy * cz;

      // point velocity: xd + omega x cpt
      float vx = xdx + (omy * cz - omz * cy);
      float vy = xdy + (omz * cx - omx * cz);
      float vz = xdz + (omx * cy - omy * cx);
      float nx = snx[n], ny = sny[n], nz = snz[n];
      float xdn = vx * nx + vy * ny + vz * nz;
      float coef = -(KST_C * sdh[n] + KD_C * xdn) * sic[n] * invden;
      float Fx = coef * nx, Fy = coef * ny, Fz = coef * nz;
      float Nmag = sqrtf(Fx * Fx + Fy * Fy + Fz * Fz);

      float ffx = 0.f, ffy = 0.f, ffz = 0.f;
#pragma unroll
      for (int i = 0; i < NJ; ++i) {
        float pmk = dpm[i * NPTS + n];
        if (pmk != 0.0f) {
          float dvx = ctrl[i] * thx - vx;
          float dvy = ctrl[i] * thy - vy;
          float dvz = ctrl[i] * thz - vz;
          float dn = dvx * nx + dvy * ny + dvz * nz;
          float tX = dvx - dn * nx, tY = dvy - dn * ny, tZ = dvz - dn * nz;
          float kf = KFR_C * Nmag * pmk;
          ffx += kf * tanhf(tX); ffy += kf * tanhf(tY); ffz += kf * tanhf(tZ);
        }
      }
      float ax = Fx + ffx, ay = Fy + ffy, az = Fz + ffz;
      size_t po = ((size_t)b * NPTS + (size_t)n) * 3;
      out[o_act + po + 0] = ax; out[o_act + po + 1] = ay; out[o_act + po + 2] = az;
      tqx += cy * az - cz * ay;
      tqy += cz * ax - cx * az;
      tqz += cx * ay - cy * ax;
      fsx += ax; fsy += ay; fsz += az;

      // new_lp: rotate joint-driven points about Y at joint positions
      float lx = lpb[n * 3 + 0], ly = lpb[n * 3 + 1], lz = lpb[n * 3 + 2];
      float bm = body[n];
      float ox = bm * lx, oy = bm * ly, oz = bm * lz;
#pragma unroll
      for (int i = 0; i < NJ; ++i) {
        float pmk = dpm[i * NPTS + n];
        if (pmk != 0.0f) {
          float ux = lx - jpx[i], uy = ly - jpy[i], uz = lz - jpz[i];
          float rx = cj[i] * ux - sj[i] * uz;
          float rz = sj[i] * ux + cj[i] * uz;
          ox += pmk * (rx + jpx[i]);
          oy += pmk * (uy + jpy[i]);
          oz += pmk * (rz + jpz[i]);
        }
      }
      out[o_lp + po + 0] = ox; out[o_lp + po + 1] = oy; out[o_lp + po + 2] = oz;
    }
    float pv[13] = {pr2, sxx, syy, szz, sxy, sxz, syz,
                    tqx, tqy, tqz, fsx, fsy, fsz};
#pragma unroll
    for (int k = 0; k < 13; ++k) {
      float s = wsum32(pv[k]);
      if (lane == 0) spart[wv][5 + k] = s;
    }
  }
  __syncthreads();

  // ---------------- Stage 5: solve + integrate + scalar outputs ------------
  if (tid == 0) {
    float a[13];
#pragma unroll
    for (int k = 0; k < 13; ++k) {
      float s = 0.f;
#pragma unroll
      for (int w = 0; w < NWAVES; ++w) s += spart[w][5 + k];
      a[k] = s;
    }
    float pr2 = a[0];
    float a00 = pr2 - a[1], a11 = pr2 - a[2], a22 = pr2 - a[3];
    float a01 = -a[4], a02 = -a[5], a12 = -a[6];
    float t0 = clampf(a[7], -TLIM_C, TLIM_C);
    float t1 = clampf(a[8], -TLIM_C, TLIM_C);
    float t2 = clampf(a[9], -TLIM_C, TLIM_C);
    // symmetric 3x3 solve via adjugate
    float c00 = a11 * a22 - a12 * a12;
    float c01 = -(a01 * a22 - a12 * a02);
    float c02 = a01 * a12 - a11 * a02;
    float det = a00 * c00 + a01 * c01 + a02 * c02;
    float idet = 1.0f / det;
    float m11 = a00 * a22 - a02 * a02;
    float m12 = -(a00 * a12 - a01 * a02);
    float m22 = a00 * a11 - a01 * a01;
    float od0 = (c00 * t0 + c01 * t1 + c02 * t2) * idet;
    float od1 = (c01 * t0 + m11 * t1 + m12 * t2) * idet;
    float od2 = (c02 * t0 + m12 * t1 + m22 * t2) * idet;

    float xdd0 = a[10] / MASS_C;
    float xdd1 = a[11] / MASS_C;
    float xdd2 = (a[12] - MASS_C * GRAV_C) / MASS_C;
    float nxd0 = xdb[0] + xdd0 * DT_C;
    float nxd1 = xdb[1] + xdd1 * DT_C;
    float nxd2 = xdb[2] + xdd2 * DT_C;
    float nx0 = xb[0] + nxd0 * DT_C;
    float nx1 = xb[1] + nxd1 * DT_C;
    float nx2 = xb[2] + nxd2 * DT_C;
    float no0 = omb[0] + od0 * DT_C;
    float no1 = omb[1] + od1 * DT_C;
    float no2 = omb[2] + od2 * DT_C;

    // exp_so3(omega_new * DT) @ R
    float w0 = no0 * DT_C, w1 = no1 * DT_C, w2 = no2 * DT_C;
    float th = sqrtf(w0 * w0 + w1 * w1 + w2 * w2);
    float thm = fmaxf(th, 1e-8f);
    float k0 = w0 / thm, k1 = w1 / thm, k2 = w2 / thm;
    float st = sinf(th), oc = 1.0f - cosf(th);
    float kk = k0 * k0 + k1 * k1 + k2 * k2;
    float E[3][3];
    E[0][0] = 1.0f + oc * (k0 * k0 - kk);
    E[0][1] = -st * k2 + oc * (k0 * k1);
    E[0][2] =  st * k1 + oc * (k0 * k2);
    E[1][0] =  st * k2 + oc * (k0 * k1);
    E[1][1] = 1.0f + oc * (k1 * k1 - kk);
    E[1][2] = -st * k0 + oc * (k1 * k2);
    E[2][0] = -st * k1 + oc * (k0 * k2);
    E[2][1] =  st * k0 + oc * (k1 * k2);
    E[2][2] = 1.0f + oc * (k2 * k2 - kk);

    out[o_x + (size_t)b * 3 + 0] = nx0;
    out[o_x + (size_t)b * 3 + 1] = nx1;
    out[o_x + (size_t)b * 3 + 2] = nx2;
    out[o_xd + (size_t)b * 3 + 0] = nxd0;
    out[o_xd + (size_t)b * 3 + 1] = nxd1;
    out[o_xd + (size_t)b * 3 + 2] = nxd2;
#pragma unroll
    for (int i = 0; i < 3; ++i)
#pragma unroll
      for (int j = 0; j < 3; ++j)
        out[o_R + (size_t)b * 9 + i * 3 + j] =
            E[i][0] * Rb[0 * 3 + j] + E[i][1] * Rb[1 * 3 + j] +
            E[i][2] * Rb[2 * 3 + j];
    out[o_om + (size_t)b * 3 + 0] = no0;
    out[o_om + (size_t)b * 3 + 1] = no1;
    out[o_om + (size_t)b * 3 + 2] = no2;
#pragma unroll
    for (int i = 0; i < NJ; ++i) {
      float tdv = clampf(ctb[NJ + i], -JVL_C, JVL_C) * DT_C;
      out[o_th + (size_t)b * NJ + i] =
          clampf(thetas[(size_t)b * NJ + i] + tdv, -JL_C, JL_C);
    }
    out[o_xdd + (size_t)b * 3 + 0] = xdd0;
    out[o_xdd + (size_t)b * 3 + 1] = xdd1;
    out[o_xdd + (size_t)b * 3 + 2] = xdd2;
    out[o_od + (size_t)b * 3 + 0] = od0;
    out[o_od + (size_t)b * 3 + 1] = od1;
    out[o_od + (size_t)b * 3 + 2] = od2;
  }
}

extern "C" void kernel_launch(void* const* d_in, const int* in_sizes, int n_in,
                              void* d_out, int out_size, void* d_ws,
                              size_t ws_size, hipStream_t stream) {
  (void)n_in; (void)out_size; (void)d_ws; (void)ws_size;
  const int B = in_sizes[0] / 3;
  dim3 grid((unsigned)B), block(BLK);
  hipLaunchKernelGGL(dphys_step, grid, block, 0, stream,
                     (const float*)d_in[0],   // x        [B,3]
                     (const float*)d_in[1],   // xd       [B,3]
                     (const float*)d_in[2],   // R        [B,3,3]
                     (const float*)d_in[3],   // lp       [B,N,3]
                     (const float*)d_in[4],   // omega    [B,3]
                     (const float*)d_in[5],   // thetas   [B,J]
                     (const float*)d_in[6],   // controls [B,2J]
                     (const float*)d_in[7],   // z_grid   [B,D,D]
                     (const float*)d_in[8],   // z_grad   [B,D,D,2]
                     (const float*)d_in[9],   // pmass    [N]
                     (const float*)d_in[10],  // part masks [J,N]
                     (const float*)d_in[11],  // body mask  [N]
                     (const float*)d_in[12],  // joint pos  [J,3]
                     (float*)d_out, B);
}